// VonMisesFisher3DLoss_58454504898727
// MI455X (gfx1250) — compile-verified
//
#include <hip/hip_runtime.h>
#include <math.h>

// Elementwise von-Mises-Fisher (m=3) NLL. 32 B HBM traffic per row ->
// bandwidth bound (~11.5 us floor at 23.3 TB/s). No matmul structure =>
// WMMA inapplicable. Round 2: collapse OCML transcendentals (663 VALU) into
// hardware v_exp/v_log/v_rcp/v_sqrt (~150 VALU) so VALU time sits under the
// HBM floor; keep b128 non-temporal loads/stores + global_prefetch_b8.

typedef float v4f __attribute__((ext_vector_type(4)));

constexpr float kLog2Pi  = 1.8378770664093453f;  // log(2*pi)
// offset = log_cmk_approx(100) - log_cmk_exact(100)
//   exact(100)  = log(100) - 100 - log1p(-exp(-200)) - log(2pi) = -97.23270688042125
//   approx(100) = -sqrt(10004)                                   = -100.01999800059987
constexpr float kHiConst = -2.7872911201786145f; // loss_hi = sqrt(4+k^2) + kHiConst - dot

__device__ __forceinline__ float vmf_loss(float dx, float dy, float dz, float kap,
                                          float tx, float ty, float tz) {
  float px = kap * dx;
  float py = kap * dy;
  float pz = kap * dz;
  float k   = __builtin_amdgcn_sqrtf(fmaf(px, px, fmaf(py, py, pz * pz)));
  float dot = fmaf(px, tx, fmaf(py, ty, pz * tz));

  // exact branch: -(log k - k - log1p(-e^{-2k}) - log 2pi)
  //             = k + log((1-e)/k) + log 2pi      (two logs fused into one)
  float e  = __expf(-2.0f * k);                       // v_exp_f32; ->0 for large k
  float q  = (1.0f - e) * __builtin_amdgcn_rcpf(k);   // v_rcp_f32
  float lo = (k + kLog2Pi) + __logf(q);               // v_log_f32

  // approx branch: sqrt(4+k^2) + offset
  float hi = __builtin_amdgcn_sqrtf(fmaf(k, k, 4.0f)) + kHiConst;

  return ((k < 100.0f) ? lo : hi) - dot;              // branchless cndmask
}

__global__ __launch_bounds__(256) void vmf3d_loss_kernel(
    const float* __restrict__ pred,   // nrows x 4  (dx,dy,dz,kappa)
    const float* __restrict__ tgt,    // nrows x 3  (unit vectors)
    float* __restrict__ out,          // nrows
    unsigned nrows) {
  const unsigned t = blockIdx.x * blockDim.x + threadIdx.x;  // 4 rows per thread
  const unsigned row0 = t * 4u;
  if (row0 >= nrows) return;

  const v4f* __restrict__ p4 = (const v4f*)pred;
  const v4f* __restrict__ t4 = (const v4f*)tgt;
  v4f* __restrict__ o4 = (v4f*)out;

  if (row0 + 4u <= nrows) {
    // ---- streaming prefetch (global_prefetch_b8), bounds-guarded ----
    const unsigned PF = 64u;            // 64 threads ahead = 256 rows ahead
    if (row0 + PF * 4u + 4u <= nrows) {
      __builtin_prefetch(&p4[row0 + PF * 4u], 0, 0);
      __builtin_prefetch(&t4[(t + PF) * 3u], 0, 0);
    }

    // ---- b128 non-temporal loads (32-bit voffsets, SADDR base) ----
    v4f pr0 = __builtin_nontemporal_load(&p4[row0 + 0u]);
    v4f pr1 = __builtin_nontemporal_load(&p4[row0 + 1u]);
    v4f pr2 = __builtin_nontemporal_load(&p4[row0 + 2u]);
    v4f pr3 = __builtin_nontemporal_load(&p4[row0 + 3u]);
    v4f ta  = __builtin_nontemporal_load(&t4[t * 3u + 0u]); // t0x t0y t0z t1x
    v4f tb  = __builtin_nontemporal_load(&t4[t * 3u + 1u]); // t1y t1z t2x t2y
    v4f tc  = __builtin_nontemporal_load(&t4[t * 3u + 2u]); // t2z t3x t3y t3z

    v4f r;
    r.x = vmf_loss(pr0.x, pr0.y, pr0.z, pr0.w, ta.x, ta.y, ta.z);
    r.y = vmf_loss(pr1.x, pr1.y, pr1.z, pr1.w, ta.w, tb.x, tb.y);
    r.z = vmf_loss(pr2.x, pr2.y, pr2.z, pr2.w, tb.z, tb.w, tc.x);
    r.w = vmf_loss(pr3.x, pr3.y, pr3.z, pr3.w, tc.y, tc.z, tc.w);

    __builtin_nontemporal_store(r, &o4[t]);
  } else {
    // scalar tail (not hit for N = 8388608, kept for robustness)
    for (unsigned r = row0; r < nrows; ++r) {
      float dx = pred[r * 4u + 0u], dy = pred[r * 4u + 1u];
      float dz = pred[r * 4u + 2u], kp = pred[r * 4u + 3u];
      float tx = tgt[r * 3u + 0u], ty = tgt[r * 3u + 1u], tz = tgt[r * 3u + 2u];
      out[r] = vmf_loss(dx, dy, dz, kp, tx, ty, tz);
    }
  }
}

extern "C" void kernel_launch(void* const* d_in, const int* in_sizes, int n_in,
                              void* d_out, int out_size, void* d_ws, size_t ws_size,
                              hipStream_t stream) {
  const float* pred = (const float*)d_in[0];   // N x 4 float32
  const float* tgt  = (const float*)d_in[1];   // N x 3 float32
  float* out = (float*)d_out;                  // N float32

  const unsigned nrows = (unsigned)(in_sizes[0] / 4);
  const unsigned nthreads = (nrows + 3u) / 4u; // 4 rows per thread
  const int block = 256;                       // 8 wave32 waves per block
  const int grid = (int)((nthreads + block - 1) / block);

  vmf3d_loss_kernel<<<grid, block, 0, stream>>>(pred, tgt, out, nrows);
}